// QueryDilatedAndGroup_63024350102238
// MI455X (gfx1250) — compile-verified
//
#include <hip/hip_runtime.h>

// Problem constants from the reference (fixed harness shapes)
#define B_   2
#define N_   16384
#define P_   4096
#define C_   64
#define S_   32          // nsample
#define RMIN2 0.16f      // 0.4^2
#define RMAX2 0.64f      // 0.8^2

typedef __attribute__((ext_vector_type(2))) float v2f;
typedef __attribute__((ext_vector_type(8))) float v8f;

static __device__ __forceinline__ unsigned ballot32(bool p) {
#if __has_builtin(__builtin_amdgcn_ballot_w32)
    return __builtin_amdgcn_ballot_w32(p);
#else
    return (unsigned)__ballot(p);
#endif
}

// One wave32 per tile of 16 query points.
// Phase 1: WMMA f32 16x16x4 computes q.p for a 16x16 (query x point) tile per
//          chunk; d2 = |q|^2 + |p|^2 - 2 q.p assembled in VALU; per-compare
//          wave ballots (ANDed in SALU) give 8 row-pair masks. A uniform
//          all-zero fast path skips selection; otherwise a
//          branchless AND/OR merge with hoisted, loop-invariant per-lane
//          select masks extracts each lane's row submask. Point loads are
//          software-pipelined one chunk ahead.
// Phase 2: gather xyz offsets + 64 feature channels; non-temporal coalesced
//          stores keep the L2-resident gather tables (8 MB features) hot.
__global__ __launch_bounds__(32) void QueryDilatedAndGroup_kernel(
    const float* __restrict__ xyz,      // (B, N, 3)
    const float* __restrict__ new_xyz,  // (B, P, 3)
    const float* __restrict__ feats,    // (B, C, N)
    float* __restrict__ out)            // (B, 3+C, P, S)
{
    __shared__ int sidx[16][S_];

    const int tile     = blockIdx.x;            // 0 .. B*(P/16)-1
    const int b        = tile / (P_ / 16);
    const int row_base = (tile % (P_ / 16)) * 16;
    const int lane     = threadIdx.x;           // 0..31 (wave32)
    const int m        = lane & 15;
    const int half     = lane >> 4;

    const float* xyzb = xyz     + (size_t)b * N_ * 3;
    const float* nxb  = new_xyz + (size_t)b * P_ * 3;

    // ---- Load query tile -> A matrix (16x4 f32) + row norms ----
    const float qx = nxb[(row_base + m) * 3 + 0];
    const float qy = nxb[(row_base + m) * 3 + 1];
    const float qz = nxb[(row_base + m) * 3 + 2];
    const float qq = qx * qx + qy * qy + qz * qz;

    v2f A;
    A.x = half ? qz  : qx;   // K=0 (lanes 0-15) / K=2 (lanes 16-31)
    A.y = half ? 0.f : qy;   // K=1 (lanes 0-15) / K=3 (lanes 16-31) -> zero pad

    // |q|^2 broadcast matching C layout: VGPR r holds rows r (lo) / r+8 (hi)
    const float qqc0 = __shfl(qq, 0 + (half << 3), 32);
    const float qqc1 = __shfl(qq, 1 + (half << 3), 32);
    const float qqc2 = __shfl(qq, 2 + (half << 3), 32);
    const float qqc3 = __shfl(qq, 3 + (half << 3), 32);
    const float qqc4 = __shfl(qq, 4 + (half << 3), 32);
    const float qqc5 = __shfl(qq, 5 + (half << 3), 32);
    const float qqc6 = __shfl(qq, 6 + (half << 3), 32);
    const float qqc7 = __shfl(qq, 7 + (half << 3), 32);

    // Loop-invariant per-lane select masks: lane m (<16) owns row m, whose
    // 16-bit submask lives in mk[m&7], half selected by m<8.
    const int      rs    = m & 7;
    const unsigned e0    = (rs == 0) ? ~0u : 0u;
    const unsigned e1    = (rs == 1) ? ~0u : 0u;
    const unsigned e2    = (rs == 2) ? ~0u : 0u;
    const unsigned e3    = (rs == 3) ? ~0u : 0u;
    const unsigned e4    = (rs == 4) ? ~0u : 0u;
    const unsigned e5    = (rs == 5) ? ~0u : 0u;
    const unsigned e6    = (rs == 6) ? ~0u : 0u;
    const unsigned e7    = (rs == 7) ? ~0u : 0u;
    const int      shift = (m < 8) ? 0 : 16;

    unsigned cnt   = 0;   // per-row sample count (lives in lanes 0..15)
    int      first = 0;   // first in-range index (fill value; 0 if none)

    // ---- Software-pipelined point scan ----
    float px = xyzb[m * 3 + 0];
    float py = xyzb[m * 3 + 1];
    float pz = xyzb[m * 3 + 2];

    for (int n0 = 0; n0 < N_; n0 += 16) {
        // Prefetch next chunk (uniform clamp -> no branch, harmless re-read on last iter)
        const int npf = (n0 + 16 < N_) ? (n0 + 16) : (N_ - 16);
        const float fx = xyzb[(npf + m) * 3 + 0];
        const float fy = xyzb[(npf + m) * 3 + 1];
        const float fz = xyzb[(npf + m) * 3 + 2];

        // ---- Current chunk -> B matrix (4x16 f32) + col norms ----
        const float pp = px * px + py * py + pz * pz;
        v2f Bm;
        Bm.x = half ? pz  : px;
        Bm.y = half ? 0.f : py;

        v8f acc = {0.f, 0.f, 0.f, 0.f, 0.f, 0.f, 0.f, 0.f};
        // D = A x B : 16x16 tile of q.p  (one V_WMMA_F32_16X16X4_F32)
        v8f dot = __builtin_amdgcn_wmma_f32_16x16x4_f32(
            /*neg_a=*/false, A, /*neg_b=*/false, Bm,
            /*c_mod=*/(short)0, acc, /*reuse_a=*/false, /*reuse_b=*/false);

        // In-range masks: mk_r bit L -> element (row r + 8*(L>=16), col L&15).
        float d2;
        d2 = qqc0 + pp - 2.f * dot[0];
        const unsigned mk0 = ballot32(d2 >= RMIN2) & ballot32(d2 < RMAX2);
        d2 = qqc1 + pp - 2.f * dot[1];
        const unsigned mk1 = ballot32(d2 >= RMIN2) & ballot32(d2 < RMAX2);
        d2 = qqc2 + pp - 2.f * dot[2];
        const unsigned mk2 = ballot32(d2 >= RMIN2) & ballot32(d2 < RMAX2);
        d2 = qqc3 + pp - 2.f * dot[3];
        const unsigned mk3 = ballot32(d2 >= RMIN2) & ballot32(d2 < RMAX2);
        d2 = qqc4 + pp - 2.f * dot[4];
        const unsigned mk4 = ballot32(d2 >= RMIN2) & ballot32(d2 < RMAX2);
        d2 = qqc5 + pp - 2.f * dot[5];
        const unsigned mk5 = ballot32(d2 >= RMIN2) & ballot32(d2 < RMAX2);
        d2 = qqc6 + pp - 2.f * dot[6];
        const unsigned mk6 = ballot32(d2 >= RMIN2) & ballot32(d2 < RMAX2);
        d2 = qqc7 + pp - 2.f * dot[7];
        const unsigned mk7 = ballot32(d2 >= RMIN2) & ballot32(d2 < RMAX2);

        // Uniform fast path: most chunks have zero in-range points
        if ((mk0 | mk1 | mk2 | mk3 | mk4 | mk5 | mk6 | mk7) != 0u) {
            // Branchless row-submask extraction (e*, shift are loop-invariant)
            const unsigned msel = (mk0 & e0) | (mk1 & e1) | (mk2 & e2) |
                                  (mk3 & e3) | (mk4 & e4) | (mk5 & e5) |
                                  (mk6 & e6) | (mk7 & e7);
            unsigned sub = (msel >> shift) & 0xFFFFu;

            if (lane < 16) {
                while (sub && cnt < S_) {
                    const int bpos = __builtin_ctz(sub);
                    const int pidx = n0 + bpos;
                    if (cnt == 0) first = pidx;
                    sidx[m][cnt] = pidx;
                    ++cnt;
                    sub &= sub - 1;
                }
            }

            // Uniform early-out once all 16 rows are full
            if (ballot32((lane < 16) ? (cnt >= S_) : true) == 0xFFFFFFFFu) break;
        }

        px = fx; py = fy; pz = fz;   // rotate pipeline
    }

    // Fill remaining slots with first found index (0 if none) — matches reference
    if (lane < 16) {
        for (unsigned k = cnt; k < S_; ++k) sidx[m][k] = first;
    }
    __syncthreads();

    // ---- Phase 2: gather + non-temporal coalesced stores. lane == slot s ----
    const float* fb   = feats + (size_t)b * C_ * N_;
    float*       outb = out   + (size_t)b * (3 + C_) * P_ * S_;

    for (int mm = 0; mm < 16; ++mm) {
        const int p = row_base + mm;
        const int j = sidx[mm][lane];

        const float cx = nxb[p * 3 + 0];
        const float cy = nxb[p * 3 + 1];
        const float cz = nxb[p * 3 + 2];

        const float gx = xyzb[j * 3 + 0] - cx;
        const float gy = xyzb[j * 3 + 1] - cy;
        const float gz = xyzb[j * 3 + 2] - cz;

        const size_t base = (size_t)p * S_ + lane;
        __builtin_nontemporal_store(gx, &outb[0 * (size_t)P_ * S_ + base]);
        __builtin_nontemporal_store(gy, &outb[1 * (size_t)P_ * S_ + base]);
        __builtin_nontemporal_store(gz, &outb[2 * (size_t)P_ * S_ + base]);

#pragma unroll 4
        for (int c = 0; c < C_; ++c) {
            const float v = fb[(size_t)c * N_ + j];
            __builtin_nontemporal_store(v, &outb[(size_t)(3 + c) * P_ * S_ + base]);
        }
    }
}

extern "C" void kernel_launch(void* const* d_in, const int* in_sizes, int n_in,
                              void* d_out, int out_size, void* d_ws, size_t ws_size,
                              hipStream_t stream) {
    (void)in_sizes; (void)n_in; (void)out_size; (void)d_ws; (void)ws_size;
    const float* xyz     = (const float*)d_in[0];  // (2,16384,3)
    const float* new_xyz = (const float*)d_in[1];  // (2,4096,3)
    const float* feats   = (const float*)d_in[2];  // (2,64,16384)
    float*       out     = (float*)d_out;          // (2,67,4096,32)

    const int tiles = B_ * (P_ / 16);              // 512 waves
    QueryDilatedAndGroup_kernel<<<tiles, 32, 0, stream>>>(xyz, new_xyz, feats, out);
}